// Neighbor_Aggregation_13451837571303
// MI455X (gfx1250) — compile-verified
//
#include <hip/hip_runtime.h>
#include <hip/hip_bf16.h>

#define N_NODES 50000
#define BATCH 4
#define N_EDGES 500000
#define H_SIZE 64

#define WAVES_PER_BLOCK 8
#define EDGES_PER_CHUNK 32
#define CHUNK_DW (EDGES_PER_CHUNK * 3)                 // 96 dwords = 384 B
#define CHUNKS_PER_BATCH (N_EDGES / EDGES_PER_CHUNK)   // 15625 (exact)
#define TOTAL_CHUNKS (BATCH * CHUNKS_PER_BATCH)        // 62500

__global__ void Neighbor_Aggregation_zero_kernel(float4* __restrict__ out, long n4) {
    long i = (long)blockIdx.x * blockDim.x + threadIdx.x;
    const long stride = (long)gridDim.x * blockDim.x;
    const float4 z = {0.0f, 0.0f, 0.0f, 0.0f};
    for (; i < n4; i += stride) out[i] = z;
}

__global__ __launch_bounds__(256) void Neighbor_Aggregation_agg_kernel(
    const float* __restrict__ H,   // (B, N, 64)
    const float* __restrict__ EW,  // (B, E, 3) = (src, dst, w) as floats
    float* __restrict__ out)       // (B, N, 64), pre-zeroed
{
    // Per-wave double-buffered staging of 32 edge triples.
    __shared__ float sedge[WAVES_PER_BLOCK * 2 * CHUNK_DW];

    const int lane  = threadIdx.x & 31;
    const int widx  = threadIdx.x >> 5;
    const int gwave = blockIdx.x * WAVES_PER_BLOCK + widx;
    const int nwav  = gridDim.x * WAVES_PER_BLOCK;

    // CDNA5 aperture rule: flat shared address low 32 bits == LDS byte offset.
    const unsigned lds_wave_base =
        (unsigned)(unsigned long long)(&sedge[widx * 2 * CHUNK_DW]);

    // Kick off async copy of one 32-edge chunk (96 dwords) into buffer `buf`.
    auto issue = [&](int chunk, int buf) {
        const int b  = chunk / CHUNKS_PER_BATCH;
        const int e0 = (chunk % CHUNKS_PER_BATCH) * EDGES_PER_CHUNK;
        const float* gp = EW + ((long)b * N_EDGES + (long)e0) * 3;
        unsigned laddr = lds_wave_base + (unsigned)(buf * CHUNK_DW * 4)
                                       + (unsigned)(lane * 4);
        unsigned long long ga = (unsigned long long)(gp + lane);
#pragma unroll
        for (int k = 0; k < 3; ++k) {
            asm volatile("global_load_async_to_lds_b32 %0, %1, off"
                         :: "v"(laddr), "v"(ga)
                         : "memory");
            laddr += 32 * 4;                         // one dword per lane
            ga    += 32 * 4;
        }
    };

    int c   = gwave;
    int buf = 0;
    if (c < TOTAL_CHUNKS) issue(c, 0);

    for (; c < TOTAL_CHUNKS; c += nwav, buf ^= 1) {
        const int cn = c + nwav;
        if (cn < TOTAL_CHUNKS) {
            issue(cn, buf ^ 1);                      // prefetch next chunk
            asm volatile("s_wait_asynccnt 0x3" ::: "memory"); // current chunk done
        } else {
            asm volatile("s_wait_asynccnt 0x0" ::: "memory");
        }

        const int b = c / CHUNKS_PER_BATCH;
        const float* __restrict__ Hb = H   + (long)b * N_NODES * H_SIZE;
        float*       __restrict__ Ob = out + (long)b * N_NODES * H_SIZE;
        const int ebase = widx * 2 * CHUNK_DW + buf * CHUNK_DW;

#pragma unroll 4
        for (int i = 0; i < EDGES_PER_CHUNK; ++i) {
            // All lanes read the same LDS address -> broadcast, conflict-free.
            const float fs = sedge[ebase + 3 * i + 0];
            const float fd = sedge[ebase + 3 * i + 1];
            const float w  = sedge[ebase + 3 * i + 2];
            const int s = (int)fs;
            const int d = (int)fd;

            // Coalesced gather: wave reads the whole 256 B row of H[b, d].
            const float2 hv =
                ((const float2*)(Hb + (long)d * H_SIZE))[lane];
            const float mx = w * hv.x;
            const float my = w * hv.y;

            // Scatter-add into out[b, s, :] via hardware f32 atomics at the
            // device-coherent point (L2). No-return form -> STOREcnt only,
            // drained by s_endpgm's implicit wait-idle.
            const unsigned long long oaddr =
                (unsigned long long)(Ob + (long)s * H_SIZE + lane * 2);
            asm volatile("global_atomic_add_f32 %0, %1, off scope:SCOPE_DEV"
                         :: "v"(oaddr), "v"(mx) : "memory");
            asm volatile("global_atomic_add_f32 %0, %1, off scope:SCOPE_DEV"
                         :: "v"(oaddr + 4ull), "v"(my) : "memory");
        }
    }
}

extern "C" void kernel_launch(void* const* d_in, const int* in_sizes, int n_in,
                              void* d_out, int out_size, void* d_ws, size_t ws_size,
                              hipStream_t stream) {
    const float* H  = (const float*)d_in[0];   // (4, 50000, 64) f32
    const float* EW = (const float*)d_in[1];   // (4, 500000, 3) f32
    float* out = (float*)d_out;                // (4, 50000, 64) f32

    const long n4 = (long)out_size / 4;        // 12.8M floats -> 3.2M float4
    Neighbor_Aggregation_zero_kernel<<<1024, 256, 0, stream>>>((float4*)out, n4);
    Neighbor_Aggregation_agg_kernel<<<2048, 256, 0, stream>>>(H, EW, out);
}